// hypergraph_part_45243185496793
// MI455X (gfx1250) — compile-verified
//
#include <hip/hip_runtime.h>
#include <hip/hip_bf16.h>
#include <math.h>

// Problem constants (from reference setup_inputs)
#define ND   2048   // diseases
#define NM   512    // medicines
#define INF_ 256    // input feature dim
#define OC   128    // output feature dim

typedef __attribute__((ext_vector_type(2))) float v2f;
typedef __attribute__((ext_vector_type(4))) float v4f;
typedef __attribute__((ext_vector_type(8))) float v8f;

static __device__ __forceinline__ float lrelu(float x) {
    return x >= 0.f ? x : 0.2f * x;
}

static __device__ __forceinline__ float waveReduceSum(float v) {
#pragma unroll
    for (int off = 16; off > 0; off >>= 1)
        v += __shfl_xor(v, off, 32);
    return v;
}

static __device__ __forceinline__ v8f wmma4(v2f a, v2f b, v8f c) {
    // (neg_a, A, neg_b, B, c_mod, C, reuse_a, reuse_b)
    return __builtin_amdgcn_wmma_f32_16x16x4_f32(false, a, false, b,
                                                 (short)0, c, false, false);
}

// ---------------------------------------------------------------------------
// Column means of c_embeddings (block 0) and m_embeddings (block 1).
// ---------------------------------------------------------------------------
__global__ void k_means(const float* __restrict__ c, const float* __restrict__ m,
                        float* __restrict__ meanC, float* __restrict__ meanM) {
    int k = threadIdx.x;  // 256 threads
    if (blockIdx.x == 0) {
        float s = 0.f;
        for (int i = 0; i < ND; ++i) s += c[(size_t)i * INF_ + k];
        meanC[k] = s * (1.0f / ND);
    } else {
        float s = 0.f;
        for (int i = 0; i < NM; ++i) s += m[(size_t)i * INF_ + k];
        meanM[k] = s * (1.0f / NM);
    }
}

// ---------------------------------------------------------------------------
// Tiny matvecs:
//   diaV[o] = W_conv[o,:] . meanC + b_conv[o]   (dia_nf broadcast row)
//   medV[o] = W_conv[o,:] . meanM + b_conv[o]   (med_nf broadcast row)
//   v[k]    = sum_o att[OC+o] * W_gat[o,k]      (so a_he = he_attr @ v)
// ---------------------------------------------------------------------------
__global__ void k_smallvecs(const float* __restrict__ Wc, const float* __restrict__ bC,
                            const float* __restrict__ Wg, const float* __restrict__ att,
                            const float* __restrict__ meanC, const float* __restrict__ meanM,
                            float* __restrict__ diaV, float* __restrict__ medV,
                            float* __restrict__ vOut) {
    int t = threadIdx.x;  // 256 threads
    if (t < OC) {
        float sd = 0.f, sm = 0.f;
        for (int k = 0; k < INF_; ++k) {
            float w = Wc[(size_t)t * INF_ + k];
            sd += w * meanC[k];
            sm += w * meanM[k];
        }
        diaV[t] = sd + bC[t];
        medV[t] = sm + bC[t];
    }
    float sv = 0.f;
    for (int o = 0; o < OC; ++o) sv += att[OC + o] * Wg[(size_t)o * INF_ + t];
    vOut[t] = sv;
}

// ---------------------------------------------------------------------------
// f32 WMMA GEMM with unit K-strides (all operand layouts arranged by their
// producers so A and B K-access is contiguous -> global_load_b64 per frag).
//   C[m,n] = sum_k A[m*sAm + k] * B[n*sBn + k]     (B given K-major per column)
// One wave computes a 32x32 tile as 2x2 16x16 sub-tiles: 4 independent
// V_WMMA_F32_16X16X4_F32 accumulator chains per 4 fragment loads per K-step.
// CT=false: store C row-major [M x N] with leading dim ldc.
// CT=true : store C transposed  [N x M] with leading dim ldc (per-lane rows
//           become contiguous -> b128 stores).
// Fragment layouts per CDNA5 ISA 7.12.2. Guard branch is wave-uniform so
// EXEC is all-ones at every WMMA.
// ---------------------------------------------------------------------------
template <bool CT>
__global__ void wmma_gemm_f32_t(const float* __restrict__ A, long sAm,
                                const float* __restrict__ B, long sBn,
                                float* __restrict__ C, int ldc,
                                int M, int N, int K) {
    int wave = blockIdx.x * (blockDim.x >> 5) + (threadIdx.x >> 5);
    int ntn = N >> 5;
    int tm = wave / ntn;
    int tn = wave - tm * ntn;
    if (tm >= (M >> 5)) return;  // wave-uniform

    int lane = threadIdx.x & 31;
    int half = lane >> 4;   // 0: K pair {0,1}, 1: K pair {2,3} of each step
    int l    = lane & 15;

    const float* pa0 = A + (long)(tm * 32 + l) * sAm + 2 * half;
    const float* pa1 = pa0 + 16 * sAm;
    const float* pb0 = B + (long)(tn * 32 + l) * sBn + 2 * half;
    const float* pb1 = pb0 + 16 * sBn;

    v8f acc00 = {}, acc01 = {}, acc10 = {}, acc11 = {};

#pragma unroll 4
    for (int kk = 0; kk < K; kk += 4) {
        v2f a0 = *(const v2f*)pa0;   // global_load_b64
        v2f a1 = *(const v2f*)pa1;
        v2f b0 = *(const v2f*)pb0;
        v2f b1 = *(const v2f*)pb1;
        acc00 = wmma4(a0, b0, acc00);
        acc01 = wmma4(a0, b1, acc01);
        acc10 = wmma4(a1, b0, acc10);
        acc11 = wmma4(a1, b1, acc11);
        pa0 += 4; pa1 += 4; pb0 += 4; pb1 += 4;
    }

    int r0 = tm * 32 + half * 8;
    int c0 = tn * 32 + l;
    if constexpr (!CT) {
#pragma unroll
        for (int r = 0; r < 8; ++r) {
            float* crow0 = C + (long)(r0 + r) * ldc;
            float* crow1 = C + (long)(r0 + r + 16) * ldc;
            crow0[c0]      = acc00[r];
            crow0[c0 + 16] = acc01[r];
            crow1[c0]      = acc10[r];
            crow1[c0 + 16] = acc11[r];
        }
    } else {
        // transposed store: element (m,n) -> C[n*ldc + m]; per accumulator the
        // 8 values sit at consecutive m -> two b128 stores per accumulator.
        float* cc0 = C + (long)c0 * ldc;         // column c0
        float* cc1 = C + (long)(c0 + 16) * ldc;  // column c0+16
        v4f t;
        t = (v4f){acc00[0], acc00[1], acc00[2], acc00[3]}; *(v4f*)(cc0 + r0)      = t;
        t = (v4f){acc00[4], acc00[5], acc00[6], acc00[7]}; *(v4f*)(cc0 + r0 + 4)  = t;
        t = (v4f){acc10[0], acc10[1], acc10[2], acc10[3]}; *(v4f*)(cc0 + r0 + 16) = t;
        t = (v4f){acc10[4], acc10[5], acc10[6], acc10[7]}; *(v4f*)(cc0 + r0 + 20) = t;
        t = (v4f){acc01[0], acc01[1], acc01[2], acc01[3]}; *(v4f*)(cc1 + r0)      = t;
        t = (v4f){acc01[4], acc01[5], acc01[6], acc01[7]}; *(v4f*)(cc1 + r0 + 4)  = t;
        t = (v4f){acc11[0], acc11[1], acc11[2], acc11[3]}; *(v4f*)(cc1 + r0 + 16) = t;
        t = (v4f){acc11[4], acc11[5], acc11[6], acc11[7]}; *(v4f*)(cc1 + r0 + 20) = t;
    }
}

// ---------------------------------------------------------------------------
// a_node[i] = x_lin[i,:] . att[:OC] for disease rows (one wave per row)
// ---------------------------------------------------------------------------
__global__ void k_anode(const float* __restrict__ xlin, const float* __restrict__ att,
                        float* __restrict__ aN) {
    int wid = blockIdx.x * (blockDim.x >> 5) + (threadIdx.x >> 5);
    int lane = threadIdx.x & 31;
    if (wid >= ND) return;
    const float* row = xlin + (size_t)wid * OC;
    float s = 0.f;
#pragma unroll
    for (int o = lane; o < OC; o += 32) s += row[o] * att[o];
    s = waveReduceSum(s);
    if (lane == 0) aN[wid] = s;
}

// ---------------------------------------------------------------------------
// a_node[ND+j] = Xmed[j,:] . att[:OC] from the transposed XmedT [OC x NM]
// (thread per medicine; loads coalesced across j).
// ---------------------------------------------------------------------------
__global__ void k_anode_med(const float* __restrict__ XmedT, const float* __restrict__ att,
                            float* __restrict__ aN) {
    int j = blockIdx.x * blockDim.x + threadIdx.x;
    if (j >= NM) return;
    float s = 0.f;
#pragma unroll 4
    for (int o = 0; o < OC; ++o) s += XmedT[(size_t)o * NM + j] * att[o];
    aN[ND + j] = s;
}

// ---------------------------------------------------------------------------
// Amax = max_j a_node[ND + j]  (leaky_relu is monotone, so the per-hyperedge
// softmax max is max(lrelu(a_d+h_d), lrelu(Amax+h_d)))
// ---------------------------------------------------------------------------
__global__ void k_amax(const float* __restrict__ aN, float* __restrict__ Amax) {
    __shared__ float sm[512];
    int t = threadIdx.x;
    sm[t] = aN[ND + t];
    __syncthreads();
    for (int off = 256; off > 0; off >>= 1) {
        if (t < off) sm[t] = fmaxf(sm[t], sm[t + off]);
        __syncthreads();
    }
    if (t == 0) Amax[0] = sm[0];
}

// ---------------------------------------------------------------------------
// a_he[d] = he_attr[d,:] . v   (one wave per row of 256)
// ---------------------------------------------------------------------------
__global__ void k_ahe(const float* __restrict__ he, const float* __restrict__ v,
                      float* __restrict__ aHe) {
    int wid = blockIdx.x * (blockDim.x >> 5) + (threadIdx.x >> 5);
    int lane = threadIdx.x & 31;
    if (wid >= ND) return;
    const float* row = he + (size_t)wid * INF_;
    float s = 0.f;
#pragma unroll
    for (int k = lane; k < INF_; k += 32) s += row[k] * v[k];
    s = waveReduceSum(s);
    if (lane == 0) aHe[wid] = s;
}

// ---------------------------------------------------------------------------
// Softmax numerators Z[d,j] (row-major, A of Z@Xmed) and Zt[j,d] (A of
// Zt-GEMM), self term zSelf[d], denominator sArr[d]. One wave per d.
// ---------------------------------------------------------------------------
__global__ void k_z(const float* __restrict__ aN, const float* __restrict__ aHe,
                    const float* __restrict__ AmaxPtr,
                    float* __restrict__ Z, float* __restrict__ Zt,
                    float* __restrict__ sArr, float* __restrict__ zSelf) {
    int d = blockIdx.x * (blockDim.x >> 5) + (threadIdx.x >> 5);
    int lane = threadIdx.x & 31;
    if (d >= ND) return;
    float ah = aHe[d];
    float Lself = lrelu(aN[d] + ah);
    float m = fmaxf(Lself, lrelu(AmaxPtr[0] + ah));
    float s = 0.f;
#pragma unroll 4
    for (int j = lane; j < NM; j += 32) {
        float z = expf(lrelu(aN[ND + j] + ah) - m);
        Z[(size_t)d * NM + j] = z;
        Zt[(size_t)j * ND + d] = z;
        s += z;
    }
    s = waveReduceSum(s);
    if (lane == 0) {
        float zs = expf(Lself - m);
        zSelf[d] = zs;
        sArr[d] = s + zs;
    }
}

// ---------------------------------------------------------------------------
// Epilogue after E = Z @ Xmed:
//   e[d]  = (Eraw[d] + zSelf[d]*x_lin[d]) / (513 * s_d)
//   fT[o*ND+d] = e / s_d                         (K-major B of the Zt GEMM)
//   out disease row d = [ (zSelf/s_d)*e + b_gat , diaV ]
// ---------------------------------------------------------------------------
__global__ void k_eps(const float* __restrict__ Eraw, const float* __restrict__ xlin,
                      const float* __restrict__ sArr, const float* __restrict__ zSelf,
                      const float* __restrict__ bG, const float* __restrict__ diaV,
                      float* __restrict__ fT, float* __restrict__ out) {
    int idx = blockIdx.x * blockDim.x + threadIdx.x;
    if (idx >= ND * OC) return;
    int d = idx >> 7;          // /OC
    int o = idx & (OC - 1);
    float sd = sArr[d], zs = zSelf[d];
    float er = Eraw[idx] + zs * xlin[idx];     // x_lin row d (ld == OC)
    float e  = er / (513.0f * sd);
    fT[(size_t)o * ND + d] = e / sd;
    size_t orow = (size_t)d * (2 * OC);
    out[orow + o]       = (zs / sd) * e + bG[o];
    out[orow + OC + o]  = diaV[o];
}

// ---------------------------------------------------------------------------
// Final medicine rows after G = Zt @ f:
//   out med row j = [ G[j]/ND + b_gat , medV ]
// ---------------------------------------------------------------------------
__global__ void k_medout(const float* __restrict__ G, const float* __restrict__ bG,
                         const float* __restrict__ medV, float* __restrict__ out) {
    int idx = blockIdx.x * blockDim.x + threadIdx.x;
    if (idx >= NM * OC) return;
    int j = idx >> 7;
    int o = idx & (OC - 1);
    size_t orow = (size_t)(ND + j) * (2 * OC);
    out[orow + o]      = G[idx] * (1.0f / ND) + bG[o];
    out[orow + OC + o] = medV[o];
}

// ---------------------------------------------------------------------------
extern "C" void kernel_launch(void* const* d_in, const int* in_sizes, int n_in,
                              void* d_out, int out_size, void* d_ws, size_t ws_size,
                              hipStream_t stream) {
    // setup_inputs order: c_it, medicine_it, c_embeddings, m_embeddings,
    //                     W_conv, b_conv, W_gat, att, b_gat, he_attr
    const float* cEmb = (const float*)d_in[2];
    const float* mEmb = (const float*)d_in[3];
    const float* Wc   = (const float*)d_in[4];
    const float* bC   = (const float*)d_in[5];
    const float* Wg   = (const float*)d_in[6];
    const float* att  = (const float*)d_in[7];
    const float* bG   = (const float*)d_in[8];
    const float* he   = (const float*)d_in[9];
    float* out = (float*)d_out;

    // Workspace carve-up (floats), ~12.1 MB total
    float* ws    = (float*)d_ws;
    size_t p = 0;
    float* xlin  = ws + p; p += (size_t)ND * OC;          // 2048 x 128 (diseases)
    float* XmedT = ws + p; p += (size_t)OC * NM;          // 128 x 512 (transposed)
    float* Z     = ws + p; p += (size_t)ND * NM;          // 2048 x 512
    float* Zt    = ws + p; p += (size_t)NM * ND;          // 512 x 2048
    float* Eraw  = ws + p; p += (size_t)ND * OC;          // 2048 x 128
    float* fT    = ws + p; p += (size_t)OC * ND;          // 128 x 2048
    float* G     = ws + p; p += (size_t)NM * OC;          // 512 x 128
    float* aN    = ws + p; p += ND + NM;
    float* aHe   = ws + p; p += ND;
    float* sArr  = ws + p; p += ND;
    float* zSelf = ws + p; p += ND;
    float* meanC = ws + p; p += INF_;
    float* meanM = ws + p; p += INF_;
    float* diaV  = ws + p; p += OC;
    float* medV  = ws + p; p += OC;
    float* vvec  = ws + p; p += INF_;
    float* Amax  = ws + p; p += 1;

    const int WPB = 4;  // waves per block in GEMM (blockDim 128)

    // 1) Column means (collapsed single-hyperedge hconvs)
    k_means<<<2, 256, 0, stream>>>(cEmb, mEmb, meanC, meanM);
    // 2) diaV/medV broadcast rows + v for a_he factorization
    k_smallvecs<<<1, 256, 0, stream>>>(Wc, bC, Wg, att, meanC, meanM, diaV, medV, vvec);

    // 3a) x_lin[diseases] = cEmb @ Wg^T   (B col n is Wg row n: sBn=IN, K contiguous)
    {
        int tiles = (ND / 32) * (OC / 32);
        wmma_gemm_f32_t<false><<<(tiles + WPB - 1) / WPB, 32 * WPB, 0, stream>>>(
            cEmb, INF_, Wg, INF_, xlin, OC, ND, OC, INF_);
    }
    // 3b) XmedT = (mEmb @ Wg^T)^T  via transposed store
    {
        int tiles = (NM / 32) * (OC / 32);
        wmma_gemm_f32_t<true><<<(tiles + WPB - 1) / WPB, 32 * WPB, 0, stream>>>(
            mEmb, INF_, Wg, INF_, XmedT, NM, NM, OC, INF_);
    }

    // 4) a_node (diseases + medicines), Amax, a_he
    k_anode<<<ND / 8, 256, 0, stream>>>(xlin, att, aN);
    k_anode_med<<<NM / 256, 256, 0, stream>>>(XmedT, att, aN);
    k_amax<<<1, 512, 0, stream>>>(aN, Amax);
    k_ahe<<<ND / 8, 256, 0, stream>>>(he, vvec, aHe);

    // 5) softmax numerators Z and Zt, denominators s, self terms
    k_z<<<ND / 8, 256, 0, stream>>>(aN, aHe, Amax, Z, Zt, sArr, zSelf);

    // 6) Eraw = Z @ Xmed  with B = XmedT (col o = XmedT row o: sBn=NM, K contig)
    {
        int tiles = (ND / 32) * (OC / 32);
        wmma_gemm_f32_t<false><<<(tiles + WPB - 1) / WPB, 32 * WPB, 0, stream>>>(
            Z, NM, XmedT, NM, Eraw, OC, ND, OC, NM);
    }
    // 7) e, fT, disease output rows
    k_eps<<<(ND * OC) / 256, 256, 0, stream>>>(Eraw, xlin, sArr, zSelf, bG, diaV,
                                               fT, out);
    // 8) G = Zt @ f  with A = Zt (sAm=ND, K contig), B = fT (sBn=ND, K contig)
    {
        int tiles = (NM / 32) * (OC / 32);
        wmma_gemm_f32_t<false><<<(tiles + WPB - 1) / WPB, 32 * WPB, 0, stream>>>(
            Zt, ND, fT, ND, G, OC, NM, OC, ND);
    }
    // 9) medicine output rows
    k_medout<<<(NM * OC) / 256, 256, 0, stream>>>(G, bG, medV, out);
}